// QuantumPatchLayer_65481071397056
// MI455X (gfx1250) — compile-verified
//
#include <hip/hip_runtime.h>
#include <hip/hip_bf16.h>
#include <stdint.h>

// QuantumPatchLayer closed form:
//   out[b, p*4 + {0,1,2,3}] = { c1*c2*c3, c0*c1, c0*c1*c2, c0*c1*c2*c3 }
// with c_i = cos(patches[b,p,i]).  rz_params provably has no effect (diagonal
// unit phases cancel in |amp|^2).  Pure streaming kernel: 16 MiB total traffic,
// HBM-bound (~0.7 us at 23.3 TB/s).  Uses CDNA5 async global->LDS 128-bit
// loads with a per-lane double buffer and s_wait_asynccnt for latency hiding.

#define BLOCK 256

__device__ __forceinline__ void async_load16(uint32_t lds_off, const void* gptr) {
    // GLOBAL_LOAD_ASYNC_TO_LDS_B128, GV mode: vdst = LDS byte offset (workgroup-
    // relative), vaddr = 64-bit global address, no SADDR ("off").
    asm volatile("global_load_async_to_lds_b128 %0, %1, off"
                 :
                 : "v"(lds_off), "v"((uint64_t)(uintptr_t)gptr)
                 : "memory");
}

__global__ __launch_bounds__(BLOCK) void qpatch_kernel(
        const float4* __restrict__ in, float4* __restrict__ out, int npatch)
{
    __shared__ float4 buf[2][BLOCK];   // 8 KiB: 2 stages x 256 lanes x 16 B

    const int  tid    = threadIdx.x;
    const long stride = (long)gridDim.x * BLOCK;
    long       idx    = (long)blockIdx.x * BLOCK + tid;

    // Flat address of a __shared__ object: high 32 bits = LDS aperture,
    // low 32 bits = workgroup-relative LDS byte offset (what the async
    // instruction's VDST expects; HW adds LDS_BASE).
    const uint32_t lds_off0 = (uint32_t)(uintptr_t)&buf[0][tid];
    const uint32_t lds_off1 = (uint32_t)(uintptr_t)&buf[1][tid];

    if (idx < npatch) async_load16(lds_off0, in + idx);

    int stage = 0;
    while (idx < npatch) {
        const long nidx     = idx + stride;
        const bool has_next = nidx < (long)npatch;

        if (has_next) {
            // Issue prefetch of next tile, then wait for the *older* one.
            // Async loads complete in order, so asynccnt<=1 guarantees the
            // current stage has landed in LDS.
            async_load16(stage ? lds_off0 : lds_off1, in + nidx);
            asm volatile("s_wait_asynccnt 1" ::: "memory");
        } else {
            asm volatile("s_wait_asynccnt 0" ::: "memory");
        }

        const float4 v = buf[stage][tid];   // ds_load_b128, own slot: no barrier

        const float c0 = __cosf(v.x);
        const float c1 = __cosf(v.y);
        const float c2 = __cosf(v.z);
        const float c3 = __cosf(v.w);

        float4 r;
        r.y = c0 * c1;            // <Z1>
        r.z = r.y * c2;           // <Z2>
        r.w = r.z * c3;           // <Z3>
        r.x = c1 * (c2 * c3);     // <Z0>

        out[idx] = r;             // global_store_b128

        idx   = nidx;
        stage ^= 1;
    }
}

extern "C" void kernel_launch(void* const* d_in, const int* in_sizes, int n_in,
                              void* d_out, int out_size, void* d_ws, size_t ws_size,
                              hipStream_t stream) {
    (void)n_in; (void)d_ws; (void)ws_size; (void)out_size;
    // d_in[0]: patches (B*P*4 float32); d_in[1]: rz_params (unused — provably
    // no effect on |amplitude|^2).
    const float4* in  = (const float4*)d_in[0];
    float4*       out = (float4*)d_out;
    const int npatch  = in_sizes[0] / 4;      // 524288 for the reference shapes

    const int iters = 4;                      // per-thread pipeline depth
    int grid = (npatch + BLOCK * iters - 1) / (BLOCK * iters);
    if (grid < 1) grid = 1;

    qpatch_kernel<<<grid, BLOCK, 0, stream>>>(in, out, npatch);
}